// GCTimeSeriesEmbedding_10548439679080
// MI455X (gfx1250) — compile-verified
//
#include <hip/hip_runtime.h>
#include <hip/hip_bf16.h>

#define DEV_INLINE __device__ __forceinline__

namespace {

constexpr int B   = 8;
constexpr int T   = 20;
constexpr int N   = 500;
constexpr int G   = 16;
constexpr int H   = 256;
constexpr int OUTF = 128;
constexpr int BT  = B * T;       // 160
constexpr int NG  = N * G;       // 8000
constexpr int H3  = 3 * H;       // 768

typedef __attribute__((ext_vector_type(2))) float v2f;
typedef __attribute__((ext_vector_type(8))) float v8f;

DEV_INLINE v2f ld2(const float* p) { return *(const v2f*)p; }

// D(16x16 f32) = A(16x4 f32) * B(4x16 f32) + C
DEV_INLINE v8f wmma4(v2f a, v2f b, v8f c) {
  return __builtin_amdgcn_wmma_f32_16x16x4_f32(
      /*neg_a=*/false, a, /*neg_b=*/false, b,
      /*c_mod=*/(short)0, c, /*reuse_a=*/false, /*reuse_b=*/false);
}

// ---------------------------------------------------------------------------
// 1) supportT[b][g][n] = input[b][n][0]*gc_w[0][g] + input[b][n][1]*gc_w[1][g]
//    (stored transposed so the gc GEMM's B-fragment loads are contiguous b64)
// ---------------------------------------------------------------------------
__global__ void support_kernel(const float* __restrict__ input,
                               const float* __restrict__ gc_w,
                               float* __restrict__ supT) {
  int idx = blockIdx.x * blockDim.x + threadIdx.x;
  if (idx >= B * N * G) return;
  int b = idx / (N * G);
  int rem = idx - b * (N * G);
  int n = rem / G;
  int g = rem - n * G;
  float v = input[(b * N + n) * 2 + 0] * gc_w[g] +
            input[(b * N + n) * 2 + 1] * gc_w[G + g];
  supT[(size_t)(b * G + g) * N + n] = v;
}

// ---------------------------------------------------------------------------
// 2) gc[bt][n][g] = adjs[bt] @ support[b] + gc_b     (HBM-bound: 160MB adjs)
//    One wave per 16-row M-tile; K loop over 500 in steps of 4 via WMMA f32.
// ---------------------------------------------------------------------------
__global__ void __launch_bounds__(128)
gc_wmma_kernel(const float* __restrict__ adjs, const float* __restrict__ supT,
               const float* __restrict__ gc_b, float* __restrict__ gc) {
  const int bt   = blockIdx.x;          // 0..159
  const int b    = bt / T;
  const int wave = threadIdx.x >> 5;
  const int lane = threadIdx.x & 31;
  const int half = lane >> 4;           // K sub-pair select
  const int r    = lane & 15;
  const int m0   = (blockIdx.y * 4 + wave) * 16;   // 32 M-tiles (last partial)
  const int row  = m0 + r;
  const int rowc = row < N ? row : (N - 1);        // clamp, keep EXEC all-1s

  const float* Arow = adjs + (size_t)bt * N * N + (size_t)rowc * N + 2 * half;
  const float* Bcol = supT + (size_t)(b * G + r) * N + 2 * half;

  v8f c = {};
#pragma unroll 4
  for (int k = 0; k < N; k += 4) {
    c = wmma4(ld2(Arow + k), ld2(Bcol + k), c);
  }

  float bias = gc_b[r];
  float* o = gc + (size_t)bt * NG;
#pragma unroll
  for (int i = 0; i < 8; ++i) {
    int m = m0 + i + 8 * half;
    if (m < N) o[m * G + r] = c[i] + bias;
  }
}

// ---------------------------------------------------------------------------
// 3) Generic row-major GEMM with bias:  Out[M,Nc] = A[M,K] @ W[Nc,K]^T + bias
//    Used for fc1 (K=8000,Nc=256) and GRU input projections (K=256,Nc=768).
//    M = 160 rows exactly (10 M-tiles). One wave per (Mtile, Ntile).
// ---------------------------------------------------------------------------
__global__ void __launch_bounds__(128)
wmma_gemm_bias_kernel(const float* __restrict__ A, const float* __restrict__ W,
                      const float* __restrict__ bias, float* __restrict__ Out,
                      int K, int Ncols) {
  const int wave = threadIdx.x >> 5;
  const int lane = threadIdx.x & 31;
  const int half = lane >> 4;
  const int r    = lane & 15;
  const int mt   = blockIdx.x;
  const int nt   = blockIdx.y * 4 + wave;
  const int row  = mt * 16 + r;
  const int col  = nt * 16 + r;

  const float* Ap = A + (size_t)row * K + 2 * half;
  const float* Wp = W + (size_t)col * K + 2 * half;

  v8f c = {};
#pragma unroll 4
  for (int k = 0; k < K; k += 4) {
    c = wmma4(ld2(Ap + k), ld2(Wp + k), c);
  }

  float bv = bias[col];
#pragma unroll
  for (int i = 0; i < 8; ++i) {
    int m = mt * 16 + i + 8 * half;
    Out[(size_t)m * Ncols + col] = c[i] + bv;
  }
}

// ---------------------------------------------------------------------------
// 4) Recurrent scan (per layer). GX = x@Wih^T+bih precomputed (kernel 3).
//    Persistent single workgroup (8 waves); per step:
//      gh = h @ Whh^T + bhh via WMMA (h fragments read from LDS),
//      gates via VALU, h kept in LDS, masked by seq_lengths.
// ---------------------------------------------------------------------------
__global__ void __launch_bounds__(256)
gru_scan_kernel(const float* __restrict__ GX, const float* __restrict__ Whh,
                const float* __restrict__ bhh, const int* __restrict__ lengths,
                float* __restrict__ xOut, float* __restrict__ hOut) {
  __shared__ float hbuf[16 * H];    // rows 8..15 stay zero (M padding)
  __shared__ float ghbuf[B * H3];

  const int tid  = threadIdx.x;
  const int wave = tid >> 5;
  const int lane = tid & 31;
  const int half = lane >> 4;
  const int r    = lane & 15;

  for (int i = tid; i < 16 * H; i += 256) hbuf[i] = 0.0f;
  __syncthreads();

  for (int t = 0; t < T; ++t) {
    // gh[0:8, 0:768] = h @ Whh^T + bhh ; 48 N-tiles, 6 per wave
    for (int q = 0; q < 6; ++q) {
      const int nt  = wave * 6 + q;
      const int col = nt * 16 + r;
      const float* Wp = Whh + (size_t)col * H + 2 * half;
      const float* Hp = hbuf + r * H + 2 * half;      // LDS ds_load_b64
      v8f c = {};
#pragma unroll 4
      for (int k = 0; k < H; k += 4) {
        c = wmma4(ld2(Hp + k), ld2(Wp + k), c);
      }
      if (half == 0) {                                // rows 0..7 = valid batch
        float bv = bhh[col];
#pragma unroll
        for (int i = 0; i < 8; ++i) ghbuf[i * H3 + col] = c[i] + bv;
      }
    }
    __syncthreads();

    // gate math: 8 batches x 256 hidden = 2048 elems, 8 per thread
    for (int e = tid; e < B * H; e += 256) {
      int bb = e >> 8;
      int j  = e & (H - 1);
      const float* gx = GX + (size_t)(bb * T + t) * H3;
      float grv = gx[j]         + ghbuf[bb * H3 + j];
      float gzv = gx[H + j]     + ghbuf[bb * H3 + H + j];
      float ghn = ghbuf[bb * H3 + 2 * H + j];
      float rg  = 1.0f / (1.0f + __expf(-grv));
      float zg  = 1.0f / (1.0f + __expf(-gzv));
      float ng  = tanhf(gx[2 * H + j] + rg * ghn);
      float hp  = hbuf[bb * H + j];
      float hn  = (1.0f - zg) * ng + zg * hp;
      hn = (t < lengths[bb]) ? hn : hp;               // pack-sequence mask
      hbuf[bb * H + j] = hn;
      xOut[(size_t)(bb * T + t) * H + j] = hn;
    }
    __syncthreads();
  }

  for (int e = tid; e < B * H; e += 256) hOut[e] = hbuf[e];
}

// ---------------------------------------------------------------------------
// 5) out = LeakyReLU_0.1( x[b, len[b]-1] @ fc2_w^T + fc2_b )   (tiny)
// ---------------------------------------------------------------------------
__global__ void fc2_kernel(const float* __restrict__ x,
                           const float* __restrict__ fc2_w,
                           const float* __restrict__ fc2_b,
                           const int* __restrict__ lengths,
                           float* __restrict__ out) {
  int idx = blockIdx.x * blockDim.x + threadIdx.x;
  if (idx >= B * OUTF) return;
  int b = idx / OUTF;
  int o = idx - b * OUTF;
  int l = lengths[b] - 1;
  const float* xr = x + (size_t)(b * T + l) * H;
  const float* wr = fc2_w + (size_t)o * H;
  float s = fc2_b[o];
#pragma unroll 4
  for (int k = 0; k < H; ++k) s += xr[k] * wr[k];
  out[idx] = s >= 0.0f ? s : 0.1f * s;
}

} // namespace

extern "C" void kernel_launch(void* const* d_in, const int* in_sizes, int n_in,
                              void* d_out, int out_size, void* d_ws, size_t ws_size,
                              hipStream_t stream) {
  const float* input = (const float*)d_in[0];   // [B,N,2]
  const float* adjs  = (const float*)d_in[1];   // [B,T,N,N]
  const int*   slen  = (const int*)  d_in[2];   // [B]
  const float* gc_w  = (const float*)d_in[3];   // [2,G]
  const float* gc_b  = (const float*)d_in[4];   // [G]
  const float* fc1_w = (const float*)d_in[5];   // [H, N*G]
  const float* fc1_b = (const float*)d_in[6];   // [H]
  const float* fc2_w = (const float*)d_in[7];   // [OUT, H]
  const float* fc2_b = (const float*)d_in[8];   // [OUT]
  const float* Wih0  = (const float*)d_in[9];   // [3H, H]
  const float* Whh0  = (const float*)d_in[10];
  const float* bih0  = (const float*)d_in[11];
  const float* bhh0  = (const float*)d_in[12];
  const float* Wih1  = (const float*)d_in[13];
  const float* Whh1  = (const float*)d_in[14];
  const float* bih1  = (const float*)d_in[15];
  const float* bhh1  = (const float*)d_in[16];

  float* out    = (float*)d_out;        // [B,OUT] = 1024 floats
  float* hidden = out + B * OUTF;       // [DEPTH,B,H] = 4096 floats

  // workspace layout (floats): ~6.2 MB total
  float* ws   = (float*)d_ws;
  float* supT = ws;                               // B*G*N      = 64000
  float* gc   = supT + (size_t)B * G * N;         // BT*N*G     = 1,280,000
  float* xA   = gc + (size_t)BT * NG;             // BT*H       = 40,960
  float* GX   = xA + (size_t)BT * H;              // BT*3H      = 122,880
  float* xB   = GX + (size_t)BT * H3;             // BT*H       = 40,960
  (void)in_sizes; (void)n_in; (void)out_size; (void)ws_size;

  const dim3 blk128(128);

  // 1) support projection (writes transposed)
  support_kernel<<<(B * N * G + 255) / 256, 256, 0, stream>>>(input, gc_w, supT);

  // 2) gc = adjs @ support + gc_b   (160 (b,t) x 8 blocks x 4 waves = 32 M-tiles)
  gc_wmma_kernel<<<dim3(BT, 8), blk128, 0, stream>>>(adjs, supT, gc_b, gc);

  // 3) x = gc @ fc1_w^T + fc1_b     ([160,8000]x[8000,256])
  wmma_gemm_bias_kernel<<<dim3(BT / 16, H / 64), blk128, 0, stream>>>(
      gc, fc1_w, fc1_b, xA, NG, H);

  // 4) GRU layer 0: GX = x@Wih^T+bih (parallel), then recurrent scan
  wmma_gemm_bias_kernel<<<dim3(BT / 16, H3 / 64), blk128, 0, stream>>>(
      xA, Wih0, bih0, GX, H, H3);
  gru_scan_kernel<<<1, 256, 0, stream>>>(GX, Whh0, bhh0, slen, xB, hidden);

  //    GRU layer 1
  wmma_gemm_bias_kernel<<<dim3(BT / 16, H3 / 64), blk128, 0, stream>>>(
      xB, Wih1, bih1, GX, H, H3);
  gru_scan_kernel<<<1, 256, 0, stream>>>(GX, Whh1, bhh1, slen, xA, hidden + B * H);

  // 5) final projection at last valid timestep + LeakyReLU(0.1)
  fc2_kernel<<<(B * OUTF + 255) / 256, 256, 0, stream>>>(xA, fc2_w, fc2_b, slen, out);
}